// LatentVariableModel_75531294867901
// MI455X (gfx1250) — compile-verified
//
#include <hip/hip_runtime.h>
#include <hip/hip_bf16.h>
#include <math.h>

typedef float v2f __attribute__((ext_vector_type(2)));
typedef float v4f __attribute__((ext_vector_type(4)));
typedef float v8f __attribute__((ext_vector_type(8)));

constexpr int Bc = 2, Nc = 1024, Tc = 4096, Hc = 256, Ec = 2, Dc = 2;

// ---------------- workspace layout (in floats) ----------------
constexpr size_t SZ_HBUF = (size_t)Bc * Nc * Tc;   // 8388608  (interleaved [B][N/4][T][4])
constexpr size_t SZ_H    = (size_t)Bc * Hc * Tc;   // 2097152
constexpr size_t OFF_HBUF = 0;
constexpr size_t OFF_H1   = OFF_HBUF + SZ_HBUF;                  // interleaved [B][H/4][T][4]
constexpr size_t OFF_H2   = OFF_H1 + SZ_H;                       // row-major [B][H][T]
constexpr size_t OFF_ZVF  = OFF_H2 + SZ_H;                       // [B,E,T,4]
constexpr size_t OFF_RFVF = OFF_ZVF + (size_t)Bc * Ec * Tc * 4;  // [E,N,4]
constexpr size_t OFF_W0P  = OFF_RFVF + (size_t)Ec * Nc * 4;      // [N]
constexpr size_t OFF_W1P  = OFF_W0P + Nc;                        // [N]
constexpr size_t OFF_AC   = OFF_W1P + Nc;                        // scalar (padded)
constexpr size_t OFF_W1I  = OFF_AC + 16;                         // [N/4][H][4]
constexpr size_t OFF_W2I  = OFF_W1I + (size_t)Nc * Hc;           // [H/4][H][4]

// ---------------- output layout (in floats) ----------------
constexpr size_t OUT_RESP = 0;
constexpr size_t OUT_Z    = OUT_RESP + (size_t)Bc * Nc * Tc;         // 8388608
constexpr size_t OUT_MU   = OUT_Z + (size_t)Bc * Tc * Ec * Dc;       // +32768
constexpr size_t OUT_LV   = OUT_MU + (size_t)Bc * Tc * Ec * Dc * 2;  // +65536

// ============ prep: zv = angle2vector(z), layout [B,E,T,4] (coalesced b128) ============
__global__ void prep_zvf_kernel(const float* __restrict__ z, float* __restrict__ zvf) {
    int tid = blockIdx.x * blockDim.x + threadIdx.x;   // B*E*T threads
    int t = tid & (Tc - 1);
    int e = (tid >> 12) & (Ec - 1);
    int b = tid >> 13;
    float z0 = z[(((size_t)b * Tc + t) * Ec + e) * Dc + 0];
    float z1 = z[(((size_t)b * Tc + t) * Ec + e) * Dc + 1];
    v4f v;
    v[0] = sinf(z0);
    v[1] = cosf(z0);
    v[2] = sinf(z1);
    v[3] = cosf(z1);
    *(v4f*)(zvf + (((size_t)(b * Ec + e) * Tc + t) << 2)) = v;
}

// ====== prep: rfv table [E,N,4] + folded combine weights + exp scale ======
__global__ void prep_rfw_kernel(const float* __restrict__ rf, const float* __restrict__ ew,
                                const float* __restrict__ fs, const float* __restrict__ ltw,
                                float* __restrict__ rfvf, float* __restrict__ w0p,
                                float* __restrict__ w1p, float* __restrict__ ac) {
    int n = blockIdx.x * blockDim.x + threadIdx.x;
    if (n >= Nc) return;
    float tw = __expf(ltw[0]);
#pragma unroll
    for (int e = 0; e < Ec; ++e) {
        float r0 = rf[((size_t)n * Ec + e) * Dc + 0];
        float r1 = rf[((size_t)n * Ec + e) * Dc + 1];
        v4f v;
        v[0] = sinf(r0);
        v[1] = cosf(r0);
        v[2] = sinf(r1);
        v[3] = cosf(r1);
        *(v4f*)(rfvf + (((size_t)e * Nc + n) << 2)) = v;
    }
    float e0 = ew[n * 2 + 0], e1 = ew[n * 2 + 1];
    float mx = fmaxf(e0, e1);
    float s0 = __expf(e0 - mx), s1 = __expf(e1 - mx);
    float base = __expf(fs[n]) * __expf(-4.0f / tw) / (s0 + s1);
    w0p[n] = base * s0;
    w1p[n] = base * s1;
    if (n == 0) ac[0] = 2.0f / tw;
}

// ============ pack weights src[M,K] row-major -> dst[K/4][M][4] ============
__global__ void pack_weights_kernel(const float* __restrict__ src, float* __restrict__ dst,
                                    int M, int K) {
    int tid = blockIdx.x * blockDim.x + threadIdx.x;  // M*K threads (coalesced dst write)
    int j = tid & 3;
    int m = (tid >> 2) % M;
    int kq = tid / (4 * M);
    dst[tid] = src[(size_t)m * K + kq * 4 + j];
}

// ===== depthwise conv1d K=9 SAME + bias, 4 channels/thread, b128 interleaved out =====
__global__ void dwconv_kernel(const float* __restrict__ x, const float* __restrict__ w,
                              const float* __restrict__ bias, float* __restrict__ h4) {
    int tid = blockIdx.x * blockDim.x + threadIdx.x;  // B*(N/4)*T
    int t = tid & (Tc - 1);
    int bn4 = tid >> 12;
    int n4 = bn4 & (Nc / 4 - 1);
    int b = bn4 >> 8;
    v4f acc;
#pragma unroll
    for (int j = 0; j < 4; ++j) {
        int n = n4 * 4 + j;
        const float* xr = x + ((size_t)b * Nc + n) * Tc;
        float s = bias[n];
#pragma unroll
        for (int k = 0; k < 9; ++k) {
            int tt = t + k - 4;
            if (tt >= 0 && tt < Tc) s += xr[tt] * w[n * 9 + k];
        }
        acc[j] = s;
    }
    *(v4f*)(h4 + ((size_t)tid << 2)) = acc;
}

// ============ fp32 WMMA GEMM + bias + relu ============
// Ai: [K/4][M][4] packed weights, Bi: [B][K/4][T][4] packed activations.
// Out: interleaved [B][M/4][T][4] (oi=1) or row-major [B][M][T] (oi=0).
// Wave tile 64(M) x 16(T); all fragment loads are single b64.
__global__ void gemm_relu_kernel(const float* __restrict__ Ai, const float* __restrict__ bias,
                                 const float* __restrict__ Bi, float* __restrict__ Out,
                                 int Kdim, int Mdim, int oi) {
    const int lane = threadIdx.x & 31;
    const int wave = threadIdx.x >> 5;
    const int lm = lane & 15;
    const int hi = lane >> 4;
    const int kx = hi << 1;  // 0 or 2: which K pair this half-wave holds
    const int t0 = (blockIdx.x * 8 + wave) * 16;
    const int m0 = blockIdx.y * 64;
    const int b = blockIdx.z;
    const int kq4 = Kdim >> 2;
    const float* Bb = Bi + ((size_t)b * kq4 * Tc << 2) + ((size_t)(t0 + lm) << 2) + kx;
    const float* Ab = Ai + ((size_t)(m0 + lm) << 2) + kx;

    v8f c0 = {}, c1 = {}, c2 = {}, c3 = {};
    for (int kq = 0; kq < kq4; ++kq) {
        v2f bf = *(const v2f*)(Bb + ((size_t)kq * Tc << 2));
        const float* Ak = Ab + ((size_t)kq * Mdim << 2);
        v2f a0 = *(const v2f*)(Ak + 0);
        v2f a1 = *(const v2f*)(Ak + 64);
        v2f a2 = *(const v2f*)(Ak + 128);
        v2f a3 = *(const v2f*)(Ak + 192);
        c0 = __builtin_amdgcn_wmma_f32_16x16x4_f32(false, a0, false, bf, (short)0, c0, false, false);
        c1 = __builtin_amdgcn_wmma_f32_16x16x4_f32(false, a1, false, bf, (short)0, c1, false, false);
        c2 = __builtin_amdgcn_wmma_f32_16x16x4_f32(false, a2, false, bf, (short)0, c2, false, false);
        c3 = __builtin_amdgcn_wmma_f32_16x16x4_f32(false, a3, false, bf, (short)0, c3, false, false);
    }
    const int t = t0 + lm;
#pragma unroll
    for (int r = 0; r < 8; ++r) {
        int mrow = r + hi * 8;
        float vals[4] = {c0[r], c1[r], c2[r], c3[r]};
#pragma unroll
        for (int ci = 0; ci < 4; ++ci) {
            int m = m0 + ci * 16 + mrow;
            float v = fmaxf(vals[ci] + bias[m], 0.f);
            if (oi) {
                Out[(((size_t)b * (Mdim >> 2) + (m >> 2)) * Tc + t) * 4 + (m & 3)] = v;
            } else {
                Out[((size_t)b * Mdim + m) * Tc + t] = v;
            }
        }
    }
}

// ============ bump readout: resp[b,n,t] = Σ_e w_e[n] * exp(a * dot4) ============
// A = rfv tile (M=n, K=4), B = zv tile (K=4, N=t): one 16x16x4 WMMA per ensemble.
__global__ void bump_kernel(const float* __restrict__ zvf, const float* __restrict__ rfvf,
                            const float* __restrict__ w0p, const float* __restrict__ w1p,
                            const float* __restrict__ ac, float* __restrict__ resp) {
    const int lane = threadIdx.x & 31;
    const int wave = threadIdx.x >> 5;
    const int lm = lane & 15;
    const int hi = lane >> 4;
    const int kx = hi << 1;
    const int t0 = (blockIdx.x * 8 + wave) * 16;
    const int n0 = blockIdx.y * 16;
    const int b = blockIdx.z;
    const float a = ac[0];

    const float* rb = rfvf + ((size_t)(n0 + lm) << 2) + kx;
    v2f A0 = *(const v2f*)(rb);
    v2f A1 = *(const v2f*)(rb + ((size_t)Nc << 2));
    const float* zb = zvf + (((size_t)b * Ec * Tc + t0 + lm) << 2) + kx;
    v2f B0 = *(const v2f*)(zb);
    v2f B1 = *(const v2f*)(zb + ((size_t)Tc << 2));

    v8f c0 = {}, c1 = {};
    c0 = __builtin_amdgcn_wmma_f32_16x16x4_f32(false, A0, false, B0, (short)0, c0, false, false);
    c1 = __builtin_amdgcn_wmma_f32_16x16x4_f32(false, A1, false, B1, (short)0, c1, false, false);

    // per-n combine weights: 4 b128 loads instead of 16 b32
    v4f w0a = *(const v4f*)(w0p + n0 + hi * 8);
    v4f w0b = *(const v4f*)(w0p + n0 + hi * 8 + 4);
    v4f w1a = *(const v4f*)(w1p + n0 + hi * 8);
    v4f w1b = *(const v4f*)(w1p + n0 + hi * 8 + 4);

    const int t = t0 + lm;
    float* rp = resp + ((size_t)b * Nc + n0 + hi * 8) * Tc + t;
#pragma unroll
    for (int r = 0; r < 8; ++r) {
        float wr0 = (r < 4) ? w0a[r & 3] : w0b[r & 3];
        float wr1 = (r < 4) ? w1a[r & 3] : w1b[r & 3];
        float v = wr0 * __expf(a * c0[r]) + wr1 * __expf(a * c1[r]);
        rp[(size_t)r * Tc] = v;
    }
}

// ============ heads: mu (normalized) + logvar + z passthrough ============
__global__ void heads_kernel(const float* __restrict__ h2, const float* __restrict__ wm,
                             const float* __restrict__ bm, const float* __restrict__ wv,
                             const float* __restrict__ bv, const float* __restrict__ z,
                             float* __restrict__ out) {
    __shared__ float sm[8 * Hc];
    __shared__ float sv[4 * Hc];
    for (int i = threadIdx.x; i < 8 * Hc; i += blockDim.x) sm[i] = wm[i];
    for (int i = threadIdx.x; i < 4 * Hc; i += blockDim.x) sv[i] = wv[i];
    __syncthreads();

    int tid = blockIdx.x * blockDim.x + threadIdx.x;  // B*T threads
    int t = tid & (Tc - 1);
    int b = tid >> 12;
    float am[8] = {0, 0, 0, 0, 0, 0, 0, 0};
    float av[4] = {0, 0, 0, 0};
    const float* hcol = h2 + (size_t)b * Hc * Tc + t;
    for (int hh = 0; hh < Hc; ++hh) {
        float v = hcol[(size_t)hh * Tc];
#pragma unroll
        for (int o = 0; o < 8; ++o) am[o] += sm[o * Hc + hh] * v;
#pragma unroll
        for (int o = 0; o < 4; ++o) av[o] += sv[o * Hc + hh] * v;
    }
    // z passthrough
#pragma unroll
    for (int i = 0; i < 4; ++i) out[OUT_Z + (size_t)tid * 4 + i] = z[(size_t)tid * 4 + i];
    // mu normalized on unit circle + logvar
#pragma unroll
    for (int p = 0; p < 4; ++p) {
        float m0 = am[2 * p] + bm[2 * p];
        float m1 = am[2 * p + 1] + bm[2 * p + 1];
        float inv = rsqrtf(m0 * m0 + m1 * m1);
        out[OUT_MU + ((size_t)tid * 4 + p) * 2 + 0] = m0 * inv;
        out[OUT_MU + ((size_t)tid * 4 + p) * 2 + 1] = m1 * inv;
        out[OUT_LV + (size_t)tid * 4 + p] = av[p] + bv[p];
    }
}

extern "C" void kernel_launch(void* const* d_in, const int* in_sizes, int n_in,
                              void* d_out, int out_size, void* d_ws, size_t ws_size,
                              hipStream_t stream) {
    (void)in_sizes; (void)n_in; (void)out_size; (void)ws_size;
    const float* x    = (const float*)d_in[0];
    const float* z    = (const float*)d_in[1];
    const float* dw_w = (const float*)d_in[2];
    const float* dw_b = (const float*)d_in[3];
    const float* w1   = (const float*)d_in[4];
    const float* b1   = (const float*)d_in[5];
    const float* w2   = (const float*)d_in[6];
    const float* b2   = (const float*)d_in[7];
    const float* wm   = (const float*)d_in[8];
    const float* bm   = (const float*)d_in[9];
    const float* wv   = (const float*)d_in[10];
    const float* bv   = (const float*)d_in[11];
    const float* rf   = (const float*)d_in[12];
    const float* ewr  = (const float*)d_in[13];
    const float* fs   = (const float*)d_in[14];
    const float* ltw  = (const float*)d_in[15];
    float* out = (float*)d_out;
    float* ws = (float*)d_ws;

    float* hbuf = ws + OFF_HBUF;
    float* h1   = ws + OFF_H1;
    float* h2   = ws + OFF_H2;
    float* zvf  = ws + OFF_ZVF;
    float* rfvf = ws + OFF_RFVF;
    float* w0p  = ws + OFF_W0P;
    float* w1p  = ws + OFF_W1P;
    float* ac   = ws + OFF_AC;
    float* w1i  = ws + OFF_W1I;
    float* w2i  = ws + OFF_W2I;

    // ---- bump path prep ----
    prep_zvf_kernel<<<(Bc * Ec * Tc) / 256, 256, 0, stream>>>(z, zvf);
    prep_rfw_kernel<<<Nc / 256, 256, 0, stream>>>(rf, ewr, fs, ltw, rfvf, w0p, w1p, ac);

    // ---- bump readout (WMMA 16x16x4 f32) ----
    dim3 gb(Tc / 128, Nc / 16, Bc);
    bump_kernel<<<gb, 256, 0, stream>>>(zvf, rfvf, w0p, w1p, ac, out + OUT_RESP);

    // ---- encoder path ----
    pack_weights_kernel<<<(Hc * Nc) / 256, 256, 0, stream>>>(w1, w1i, Hc, Nc);
    pack_weights_kernel<<<(Hc * Hc) / 256, 256, 0, stream>>>(w2, w2i, Hc, Hc);
    dwconv_kernel<<<(Bc * (Nc / 4) * Tc) / 256, 256, 0, stream>>>(x, dw_w, dw_b, hbuf);

    dim3 gg(Tc / 128, Hc / 64, Bc);
    gemm_relu_kernel<<<gg, 256, 0, stream>>>(w1i, b1, hbuf, h1, Nc, Hc, 1);
    gemm_relu_kernel<<<gg, 256, 0, stream>>>(w2i, b2, h1, h2, Hc, Hc, 0);

    heads_kernel<<<(Bc * Tc) / 256, 256, 0, stream>>>(h2, wm, bm, wv, bv, z, out);
}